// TemporalDualBranch_5652176962347
// MI455X (gfx1250) — compile-verified
//
#include <hip/hip_runtime.h>
#include <math.h>

// Problem constants (from reference)
#define BB 4
#define TT 256
#define NN 64
#define DD 128
#define HH 4
#define HD 32
#define RR (BB * NN * TT)   // 65536 rows

// GEMM blocking
#define BM 128   // block tile rows
#define BN 64    // block tile cols
#define KC 64    // K chunk staged in LDS
#define BL 128   // threads per block (4 waves)

typedef __attribute__((ext_vector_type(2))) float v2f;
typedef __attribute__((ext_vector_type(8))) float v8f;

// ---------------------------------------------------------------------------
// CDNA5 async global->LDS copy (ASYNCcnt-tracked DMA), with sync fallback.
// Builtin signature (from hipcc diagnostic): (int4 AS1*, int4 AS3*, Ii, Ii).
// ---------------------------------------------------------------------------
#if defined(__has_builtin)
#if __has_builtin(__builtin_amdgcn_global_load_async_to_lds_b128) && \
    __has_builtin(__builtin_amdgcn_s_wait_asynccnt)
#define HAVE_ASYNC_LDS 1
#endif
#endif

#ifdef HAVE_ASYNC_LDS
typedef int vi4 __attribute__((__vector_size__(4 * sizeof(int))));
typedef __attribute__((address_space(1))) vi4 as1_vi4;
typedef __attribute__((address_space(3))) vi4 as3_vi4;
#endif

__device__ __forceinline__ void copy16(const float* g, float* l) {
#ifdef HAVE_ASYNC_LDS
  as1_vi4* gp = reinterpret_cast<as1_vi4*>((uintptr_t)g);
  as3_vi4* lp = reinterpret_cast<as3_vi4*>((uintptr_t)l);
  __builtin_amdgcn_global_load_async_to_lds_b128(gp, lp, 0, 0);
#else
  *reinterpret_cast<float4*>(l) = *reinterpret_cast<const float4*>(g);
#endif
}

__device__ __forceinline__ void stage_sync() {
#ifdef HAVE_ASYNC_LDS
  __builtin_amdgcn_s_wait_asynccnt(0);
#endif
  __syncthreads();
}

// ---------------------------------------------------------------------------
// Activation epilogues: 0=none, 1=exact GELU (erf), 2=tanh, 3=sigmoid
// ---------------------------------------------------------------------------
__device__ __forceinline__ float act_apply(float v, int act) {
  if (act == 1) return 0.5f * v * (1.0f + erff(v * 0.70710678118654752440f));
  if (act == 2) return tanhf(v);
  if (act == 3) return 1.0f / (1.0f + expf(-v));
  return v;
}

// ---------------------------------------------------------------------------
// fp32 GEMM: C[R,Nc] = act(A[R,K] @ B[K,Nc] + bias)
// Block = 128 threads = 4 waves (2x2). Block tile 128x64; wave tile 64x32
// (4 M-subtiles x 2 N-subtiles of 16x16). K staged in 64-chunks,
// double-buffered in LDS via async global->LDS DMA when available.
// Per K-step: 4 A-pairs + 2 B-pairs feed 8 V_WMMA_F32_16X16X4_F32.
// ---------------------------------------------------------------------------
__global__ void gemm_wmma_kernel(const float* __restrict__ A,
                                 const float* __restrict__ Bm,
                                 const float* __restrict__ bias,
                                 float* __restrict__ C,
                                 int K, int Nc, int act) {
  extern __shared__ float smem[];
  float* sAb = smem;                 // [2][BM*KC]
  float* sBb = smem + 2 * BM * KC;   // [2][KC*BN]

  const int tid = threadIdx.x;
  const int rowBase = blockIdx.x * BM;
  const int colBase = blockIdx.y * BN;
  const int nk = K / KC;

  // Stage K-chunk kb into LDS buffer buf (async DMA when available).
  auto stage = [&](int kb, int buf) {
    float* dA = sAb + buf * (BM * KC);
    float* dB = sBb + buf * (KC * BN);
    const float* gA = A + (size_t)rowBase * K + kb * KC;
    const float* gB = Bm + (size_t)(kb * KC) * Nc + colBase;
#pragma unroll
    for (int j = 0; j < (BM * KC) / (4 * BL); ++j) {  // 16 x b128 per thread
      int idx = tid + j * BL;
      int r = idx >> 4;
      int cs = (idx & 15) << 2;
      copy16(gA + (size_t)r * K + cs, dA + r * KC + cs);
    }
#pragma unroll
    for (int j = 0; j < (KC * BN) / (4 * BL); ++j) {  // 8 x b128 per thread
      int idx = tid + j * BL;
      int r = idx >> 4;
      int cs = (idx & 15) << 2;
      copy16(gB + (size_t)r * Nc + cs, dB + r * BN + cs);
    }
  };

  const int lane = tid & 31;
  const int wv = tid >> 5;
  const int mw = (wv & 1) * 64;    // wave row offset in block tile
  const int nw = (wv >> 1) * 32;   // wave col offset in block tile
  const int m = lane & 15;         // A row / C column within 16x16 subtile
  const int hi = lane >> 4;        // K-pair selector (lanes 16-31: K+2,K+3)

  v8f acc[4][2] = {};

  stage(0, 0);
  stage_sync();

  for (int kb = 0; kb < nk; ++kb) {
    const int cur = kb & 1;
    if (kb + 1 < nk) stage(kb + 1, cur ^ 1);
    if (kb + 2 < nk)  // warm L2 for the chunk after next (global_prefetch_b8)
      __builtin_prefetch(A + (size_t)(rowBase + (tid >> 1)) * K + (kb + 2) * KC, 0, 0);

    const float* cA = sAb + cur * (BM * KC);
    const float* cB = sBb + cur * (KC * BN);

#pragma unroll 4
    for (int k0 = 0; k0 < KC; k0 += 4) {
      const int ka = k0 + 2 * hi;
      v2f av[4], bv[2];
#pragma unroll
      for (int mt = 0; mt < 4; ++mt) {
        const float* p = cA + (size_t)(mw + mt * 16 + m) * KC + ka;
        av[mt].x = p[0];
        av[mt].y = p[1];
      }
#pragma unroll
      for (int nt = 0; nt < 2; ++nt) {
        const float* p = cB + (size_t)ka * BN + nw + nt * 16 + m;
        bv[nt].x = p[0];
        bv[nt].y = p[BN];
      }
#pragma unroll
      for (int mt = 0; mt < 4; ++mt)
#pragma unroll
        for (int nt = 0; nt < 2; ++nt)
          acc[mt][nt] = __builtin_amdgcn_wmma_f32_16x16x4_f32(
              false, av[mt], false, bv[nt], (short)0, acc[mt][nt], false, false);
    }
    stage_sync();
  }

  // C/D layout: VGPR r2 holds M=r2 (lanes 0-15) or M=r2+8 (lanes 16-31)
#pragma unroll
  for (int nt = 0; nt < 2; ++nt) {
    const int col = colBase + nw + nt * 16 + m;
    const float bb = bias[col];
#pragma unroll
    for (int mt = 0; mt < 4; ++mt) {
#pragma unroll
      for (int r2 = 0; r2 < 8; ++r2) {
        int row = rowBase + mw + mt * 16 + hi * 8 + r2;
        C[(size_t)row * Nc + col] = act_apply(acc[mt][nt][r2] + bb, act);
      }
    }
  }
}

// ---------------------------------------------------------------------------
// prep_local: xl[bn,t,:] = x[b,t,n,:] - softmax(window_weight)-weighted
// 3-tap trend (zero-padded in time). One thread per float4.
// ---------------------------------------------------------------------------
__global__ void prep_local_kernel(const float* __restrict__ x,
                                  const float* __restrict__ wwraw,
                                  float* __restrict__ xl) {
  int idx = blockIdx.x * blockDim.x + threadIdx.x;  // RR*32 vec4s
  int d4 = idx & 31;
  int n = (idx >> 5) & 63;
  int t = (idx >> 11) & 255;
  int b = idx >> 19;

  float w0 = wwraw[0], w1 = wwraw[1], w2 = wwraw[2];
  float mw = fmaxf(w0, fmaxf(w1, w2));
  float e0 = expf(w0 - mw), e1 = expf(w1 - mw), e2 = expf(w2 - mw);
  float inv = 1.0f / (e0 + e1 + e2);
  e0 *= inv; e1 *= inv; e2 *= inv;

  const float4* X4 = reinterpret_cast<const float4*>(x);
  size_t base = ((size_t)(b * TT + t) * NN + n) * 32 + d4;
  float4 xc = X4[base];
  float4 xm = make_float4(0.f, 0.f, 0.f, 0.f);
  float4 xp = make_float4(0.f, 0.f, 0.f, 0.f);
  if (t > 0)      xm = X4[base - (size_t)NN * 32];
  if (t < TT - 1) xp = X4[base + (size_t)NN * 32];

  float4 o;
  o.x = xc.x - (e0 * xm.x + e1 * xc.x + e2 * xp.x);
  o.y = xc.y - (e0 * xm.y + e1 * xc.y + e2 * xp.y);
  o.z = xc.z - (e0 * xm.z + e1 * xc.z + e2 * xp.z);
  o.w = xc.w - (e0 * xm.w + e1 * xc.w + e2 * xp.w);

  int r = (b * NN + n) * TT + t;
  reinterpret_cast<float4*>(xl)[(size_t)r * 32 + d4] = o;
}

// ---------------------------------------------------------------------------
// Banded (|i-j|<=1) multi-head attention: off-band -1e9 scores underflow to
// exactly 0 after fp32 softmax, so only 3 neighbors matter.
// One thread per (bn, t, h).
// ---------------------------------------------------------------------------
__global__ void attn_kernel(const float* __restrict__ qkv,
                            float* __restrict__ outp) {
  int tid = blockIdx.x * blockDim.x + threadIdx.x;  // RR*HH threads
  int h = tid & 3;
  int r = tid >> 2;
  int t = r & 255;
  int bn = r >> 8;

  const float* qp = qkv + (size_t)r * 384 + h * 32;
  float q[32];
#pragma unroll
  for (int i = 0; i < 32; ++i) q[i] = qp[i];

  float s[3];
  float mmax = -3.0e38f;
#pragma unroll
  for (int j0 = 0; j0 < 3; ++j0) {
    int j = t + j0 - 1;
    if (j >= 0 && j < TT) {
      const float* kp = qkv + ((size_t)(bn * TT + j)) * 384 + 128 + h * 32;
      float d = 0.f;
#pragma unroll
      for (int i = 0; i < 32; ++i) d += q[i] * kp[i];
      s[j0] = d * 0.17677669529663687f;  // 1/sqrt(32)
      mmax = fmaxf(mmax, s[j0]);
    } else {
      s[j0] = -3.0e38f;
    }
  }

  float o[32];
#pragma unroll
  for (int i = 0; i < 32; ++i) o[i] = 0.f;
  float den = 0.f;
#pragma unroll
  for (int j0 = 0; j0 < 3; ++j0) {
    int j = t + j0 - 1;
    if (j >= 0 && j < TT) {
      float p = expf(s[j0] - mmax);
      den += p;
      const float* vp = qkv + ((size_t)(bn * TT + j)) * 384 + 256 + h * 32;
#pragma unroll
      for (int i = 0; i < 32; ++i) o[i] += p * vp[i];
    }
  }
  float invd = 1.0f / den;
  float* op = outp + (size_t)r * 128 + h * 32;
#pragma unroll
  for (int i = 0; i < 32; ++i) op[i] = o[i] * invd;
}

// ---------------------------------------------------------------------------
// LayerNorm over D=128, one wave32 per row (4 floats/lane), shfl_xor reduce.
// ---------------------------------------------------------------------------
__global__ void ln_kernel(float* __restrict__ X,
                          const float* __restrict__ g,
                          const float* __restrict__ b) {
  int row = blockIdx.x * 8 + (threadIdx.x >> 5);
  int lane = threadIdx.x & 31;
  float4* X4 = reinterpret_cast<float4*>(X);
  float4 v = X4[(size_t)row * 32 + lane];

  float s = v.x + v.y + v.z + v.w;
  float sq = v.x * v.x + v.y * v.y + v.z * v.z + v.w * v.w;
#pragma unroll
  for (int off = 16; off > 0; off >>= 1) {
    s += __shfl_xor(s, off, 32);
    sq += __shfl_xor(sq, off, 32);
  }
  float mean = s * (1.0f / 128.0f);
  float var = sq * (1.0f / 128.0f) - mean * mean;
  float inv = rsqrtf(var + 1e-5f);

  const float4 g4 = reinterpret_cast<const float4*>(g)[lane];
  const float4 b4 = reinterpret_cast<const float4*>(b)[lane];
  float4 o;
  o.x = (v.x - mean) * inv * g4.x + b4.x;
  o.y = (v.y - mean) * inv * g4.y + b4.y;
  o.z = (v.z - mean) * inv * g4.z + b4.z;
  o.w = (v.w - mean) * inv * g4.w + b4.w;
  X4[(size_t)row * 32 + lane] = o;
}

// ---------------------------------------------------------------------------
// prep_conv: x_pos = x + pos_enc; depthwise conv3 (pad 1, zeros AFTER pos add
// per reference) + conv bias. Output layout [bn,t,D]. One thread per float4.
// ---------------------------------------------------------------------------
__global__ void prep_conv_kernel(const float* __restrict__ x,
                                 const float* __restrict__ pos,
                                 const float* __restrict__ cw,
                                 const float* __restrict__ cb,
                                 float* __restrict__ xconv) {
  int idx = blockIdx.x * blockDim.x + threadIdx.x;  // RR*32 vec4s
  int d4 = idx & 31;
  int n = (idx >> 5) & 63;
  int t = (idx >> 11) & 255;
  int b = idx >> 19;
  int d0 = d4 * 4;

  const float4* X4 = reinterpret_cast<const float4*>(x);
  const float4* P4 = reinterpret_cast<const float4*>(pos);
  size_t base = ((size_t)(b * TT + t) * NN + n) * 32 + d4;

  float xm[4] = {0.f, 0.f, 0.f, 0.f}, xc[4], xp[4] = {0.f, 0.f, 0.f, 0.f};
  {
    float4 v = X4[base];
    float4 p = P4[(size_t)t * 32 + d4];
    xc[0] = v.x + p.x; xc[1] = v.y + p.y; xc[2] = v.z + p.z; xc[3] = v.w + p.w;
  }
  if (t > 0) {
    float4 v = X4[base - (size_t)NN * 32];
    float4 p = P4[(size_t)(t - 1) * 32 + d4];
    xm[0] = v.x + p.x; xm[1] = v.y + p.y; xm[2] = v.z + p.z; xm[3] = v.w + p.w;
  }
  if (t < TT - 1) {
    float4 v = X4[base + (size_t)NN * 32];
    float4 p = P4[(size_t)(t + 1) * 32 + d4];
    xp[0] = v.x + p.x; xp[1] = v.y + p.y; xp[2] = v.z + p.z; xp[3] = v.w + p.w;
  }

  float o[4];
#pragma unroll
  for (int c = 0; c < 4; ++c) {
    int d = d0 + c;
    float w0 = cw[d * 3 + 0], w1 = cw[d * 3 + 1], w2 = cw[d * 3 + 2];
    o[c] = w0 * xm[c] + w1 * xc[c] + w2 * xp[c] + cb[d];
  }
  int r = (b * NN + n) * TT + t;
  reinterpret_cast<float4*>(xconv)[(size_t)r * 32 + d4] =
      make_float4(o[0], o[1], o[2], o[3]);
}

// ---------------------------------------------------------------------------
// Mixed pool: k=6, stride 1, pad (2,2) -> length T-1, right zero-pad to T.
// avg divides by 6 always (zero pad contributes 0); max over valid entries.
// ---------------------------------------------------------------------------
__global__ void pool_kernel(const float* __restrict__ xconv,
                            const float* __restrict__ alpha_p,
                            float* __restrict__ outp) {
  int idx = blockIdx.x * blockDim.x + threadIdx.x;  // RR*32 vec4s
  int d4 = idx & 31;
  int r = idx >> 5;
  int t = r & 255;
  int bn = r >> 8;

  float4 res = make_float4(0.f, 0.f, 0.f, 0.f);
  if (t < TT - 1) {
    float a = 1.0f / (1.0f + expf(-alpha_p[0]));
    const float4* XC4 = reinterpret_cast<const float4*>(xconv);
    float sx = 0.f, sy = 0.f, sz = 0.f, sw = 0.f;
    float mx = -3.0e38f, my = -3.0e38f, mz = -3.0e38f, mw = -3.0e38f;
    for (int j = t - 2; j <= t + 3; ++j) {
      if (j >= 0 && j < TT) {
        float4 v = XC4[((size_t)(bn * TT + j)) * 32 + d4];
        sx += v.x; sy += v.y; sz += v.z; sw += v.w;
        mx = fmaxf(mx, v.x); my = fmaxf(my, v.y);
        mz = fmaxf(mz, v.z); mw = fmaxf(mw, v.w);
      }
    }
    const float inv6 = 1.0f / 6.0f;
    res.x = a * (sx * inv6) + (1.f - a) * mx;
    res.y = a * (sy * inv6) + (1.f - a) * my;
    res.z = a * (sz * inv6) + (1.f - a) * mz;
    res.w = a * (sw * inv6) + (1.f - a) * mw;
  }
  reinterpret_cast<float4*>(outp)[(size_t)r * 32 + d4] = res;
}

// ---------------------------------------------------------------------------
// concat [x_loc | x_glob] -> [R, 256]
// ---------------------------------------------------------------------------
__global__ void concat_kernel(const float* __restrict__ xloc,
                              const float* __restrict__ xglob,
                              float* __restrict__ cat) {
  int idx = blockIdx.x * blockDim.x + threadIdx.x;  // RR*64 vec4s
  int c4 = idx & 63;
  int r = idx >> 6;
  float4 v;
  if (c4 < 32)
    v = reinterpret_cast<const float4*>(xloc)[(size_t)r * 32 + c4];
  else
    v = reinterpret_cast<const float4*>(xglob)[(size_t)r * 32 + (c4 - 32)];
  reinterpret_cast<float4*>(cat)[(size_t)r * 64 + c4] = v;
}

// ---------------------------------------------------------------------------
// Final: out = x + softplus(res_scale) * (gate*x_loc + (1-gate)*x_glob)
// with [bn,t,D] -> [B,T,N,D] relayout.
// ---------------------------------------------------------------------------
__device__ __forceinline__ float softplus_f(float v) {
  return (v > 0.f) ? (v + log1pf(expf(-v))) : log1pf(expf(v));
}

__global__ void final_kernel(const float* __restrict__ x,
                             const float* __restrict__ gate,
                             const float* __restrict__ xloc,
                             const float* __restrict__ xglob,
                             const float* __restrict__ res_scale,
                             float* __restrict__ out) {
  int idx = blockIdx.x * blockDim.x + threadIdx.x;  // RR*32 vec4s
  int d4 = idx & 31;
  int n = (idx >> 5) & 63;
  int t = (idx >> 11) & 255;
  int b = idx >> 19;
  int r = (b * NN + n) * TT + t;

  float4 xv = reinterpret_cast<const float4*>(x)[((size_t)(b * TT + t) * NN + n) * 32 + d4];
  float4 g  = reinterpret_cast<const float4*>(gate)[(size_t)r * 32 + d4];
  float4 xl = reinterpret_cast<const float4*>(xloc)[(size_t)r * 32 + d4];
  float4 xg = reinterpret_cast<const float4*>(xglob)[(size_t)r * 32 + d4];
  float4 rs = reinterpret_cast<const float4*>(res_scale)[d4];

  float4 o;
  o.x = xv.x + softplus_f(rs.x) * (g.x * xl.x + (1.f - g.x) * xg.x);
  o.y = xv.y + softplus_f(rs.y) * (g.y * xl.y + (1.f - g.y) * xg.y);
  o.z = xv.z + softplus_f(rs.z) * (g.z * xl.z + (1.f - g.z) * xg.z);
  o.w = xv.w + softplus_f(rs.w) * (g.w * xl.w + (1.f - g.w) * xg.w);
  reinterpret_cast<float4*>(out)[((size_t)(b * TT + t) * NN + n) * 32 + d4] = o;
}

// ---------------------------------------------------------------------------
// Host launcher
// ---------------------------------------------------------------------------
extern "C" void kernel_launch(void* const* d_in, const int* in_sizes, int n_in,
                              void* d_out, int out_size, void* d_ws, size_t ws_size,
                              hipStream_t stream) {
  const float* x          = (const float*)d_in[0];
  const float* window_w   = (const float*)d_in[1];
  const float* local_w1   = (const float*)d_in[2];
  const float* local_b1   = (const float*)d_in[3];
  const float* local_w2   = (const float*)d_in[4];
  const float* local_b2   = (const float*)d_in[5];
  const float* local_ln_g = (const float*)d_in[6];
  const float* local_ln_b = (const float*)d_in[7];
  const float* qkv_w      = (const float*)d_in[8];
  const float* qkv_b      = (const float*)d_in[9];
  const float* out_w      = (const float*)d_in[10];
  const float* out_b      = (const float*)d_in[11];
  const float* conv_w     = (const float*)d_in[12];
  const float* conv_b     = (const float*)d_in[13];
  const float* alpha      = (const float*)d_in[14];
  const float* pos_enc    = (const float*)d_in[15];
  const float* global_w1  = (const float*)d_in[16];
  const float* global_b1  = (const float*)d_in[17];
  const float* global_w2  = (const float*)d_in[18];
  const float* global_b2  = (const float*)d_in[19];
  const float* global_ln_g= (const float*)d_in[20];
  const float* global_ln_b= (const float*)d_in[21];
  const float* gate_w1    = (const float*)d_in[22];
  const float* gate_b1    = (const float*)d_in[23];
  const float* gate_w2    = (const float*)d_in[24];
  const float* gate_b2    = (const float*)d_in[25];
  const float* res_scale  = (const float*)d_in[26];
  float* out = (float*)d_out;

  const size_t R = (size_t)RR;
  float* ws = (float*)d_ws;
  float* P0 = ws;             // R*128: xl -> attn_out -> x_loc
  float* P1 = ws + R * 128;   // R*384: qkv -> proj/h1/cat/gate
  float* P2 = ws + R * 512;   // R*128: xconv -> x_glob
  float* P3 = ws + R * 640;   // R*128: x_glob_in -> tanh_out
  float* h1   = P1 + R * 128; // R*256 region inside P1
  float* gate = P1 + R * 256; // R*128 region inside P1

  const int EW_BLOCKS = (RR * 32) / 256;               // elementwise vec4 grids
  const size_t GEMM_LDS = (2 * BM * KC + 2 * KC * BN) * sizeof(float);  // 96 KB

  // local branch
  prep_local_kernel<<<EW_BLOCKS, 256, 0, stream>>>(x, window_w, P0);
  gemm_wmma_kernel<<<dim3(RR / BM, 384 / BN), BL, GEMM_LDS, stream>>>(
      P0, qkv_w, qkv_b, P1, 128, 384, 0);
  attn_kernel<<<(RR * HH) / 256, 256, 0, stream>>>(P1, P0);
  gemm_wmma_kernel<<<dim3(RR / BM, 128 / BN), BL, GEMM_LDS, stream>>>(
      P0, out_w, out_b, P1, 128, 128, 0);
  gemm_wmma_kernel<<<dim3(RR / BM, 256 / BN), BL, GEMM_LDS, stream>>>(
      P1, local_w1, local_b1, h1, 128, 256, 1 /*gelu*/);
  gemm_wmma_kernel<<<dim3(RR / BM, 128 / BN), BL, GEMM_LDS, stream>>>(
      h1, local_w2, local_b2, P0, 256, 128, 0);
  ln_kernel<<<RR / 8, 256, 0, stream>>>(P0, local_ln_g, local_ln_b);  // x_loc = P0

  // global branch
  prep_conv_kernel<<<EW_BLOCKS, 256, 0, stream>>>(x, pos_enc, conv_w, conv_b, P2);
  pool_kernel<<<EW_BLOCKS, 256, 0, stream>>>(P2, alpha, P3);
  gemm_wmma_kernel<<<dim3(RR / BM, 256 / BN), BL, GEMM_LDS, stream>>>(
      P3, global_w1, global_b1, h1, 128, 256, 1 /*gelu*/);
  gemm_wmma_kernel<<<dim3(RR / BM, 128 / BN), BL, GEMM_LDS, stream>>>(
      h1, global_w2, global_b2, P2, 256, 128, 0);
  ln_kernel<<<RR / 8, 256, 0, stream>>>(P2, global_ln_g, global_ln_b);  // x_glob = P2

  // gated fusion
  concat_kernel<<<(RR * 64) / 256, 256, 0, stream>>>(P0, P2, P1);  // cat in P1[0:R*256]
  gemm_wmma_kernel<<<dim3(RR / BM, 128 / BN), BL, GEMM_LDS, stream>>>(
      P1, gate_w1, gate_b1, P3, 256, 128, 2 /*tanh*/);
  gemm_wmma_kernel<<<dim3(RR / BM, 128 / BN), BL, GEMM_LDS, stream>>>(
      P3, gate_w2, gate_b2, gate, 128, 128, 3 /*sigmoid*/);

  final_kernel<<<EW_BLOCKS, 256, 0, stream>>>(x, gate, P0, P2, res_scale, out);
}